// SlidingWindowMemory_72627896975940
// MI455X (gfx1250) — compile-verified
//
#include <hip/hip_runtime.h>
#include <hip/hip_bf16.h>
#include <stdint.h>

// Problem shape (from reference): T=2048 steps, L=64 window, D=1024 features.
#define T_STEPS 2048
#define L_WIN   64
#define D_FEAT  1024

// Tail broadcast tiling: 63 tail rows = 7 tiles x 9 rows.
#define TILE_ROWS 9
#define N_TILES   7
#define TB        16                       // timesteps handled per block
#define THREADS   256
#define TILE_F4   (TILE_ROWS * D_FEAT / 4) // 2304 float4 per tile
#define F4_PER_T  (TILE_F4 / THREADS)      // 9 float4 per thread

// Native 4-float vector (usable with __builtin_nontemporal_store, unlike
// HIP's float4 class type).
typedef float v4f __attribute__((ext_vector_type(4)));

// ---------------------------------------------------------------------------
// Kernel A: broadcast memory[1:] (the tail) to out[d, 1:, :] for all d.
// Stage the tile in LDS via async global->LDS loads, then stream it out with
// async LDS->global stores (non-temporal: the 512MiB output exceeds L2).
// ---------------------------------------------------------------------------
__global__ __launch_bounds__(THREADS) void tail_broadcast_kernel(
    const float* __restrict__ memory, float* __restrict__ out) {
  __shared__ float lds_tile[TILE_ROWS * D_FEAT];  // 36 KiB

  const int tile = blockIdx.x % N_TILES;
  const int dgrp = blockIdx.x / N_TILES;
  const int tid  = threadIdx.x;

  // Byte offset of this tile's first element inside `memory` (skip row 0).
  const unsigned tileBaseBytes =
      (unsigned)((1 + TILE_ROWS * tile) * D_FEAT) * 4u;

  // Wave-relative LDS byte address = low 32 bits of the shared-aperture
  // flat address.
  const unsigned ldsBase = (unsigned)(uintptr_t)&lds_tile[0];

  // ---- Stage tile: global -> LDS (async, b128), tracked by ASYNCcnt ----
#pragma unroll
  for (int i = 0; i < F4_PER_T; ++i) {
    const unsigned f4   = (unsigned)(tid + i * THREADS);
    const unsigned gOff = tileBaseBytes + f4 * 16u;
    const unsigned lOff = ldsBase + f4 * 16u;
    asm volatile("global_load_async_to_lds_b128 %0, %1, %2"
                 :
                 : "v"(lOff), "v"(gOff), "s"(memory)
                 : "memory");
  }
  // Wait for this wave's async loads, then barrier so every wave sees the
  // whole tile (loads were split across waves).
  asm volatile("s_wait_asynccnt 0" ::: "memory");
  __syncthreads();

  // ---- Replicate tile: LDS -> global (async, b128, non-temporal) ----
  // Byte offset of this tile inside one [L, D] output slab.
  const unsigned slabOff = (unsigned)((1 + TILE_ROWS * tile) * D_FEAT) * 4u;
  const unsigned slabBytes = (unsigned)(L_WIN * D_FEAT) * 4u;

  for (int j = 0; j < TB; ++j) {
    const unsigned d     = (unsigned)(dgrp * TB + j);
    const unsigned dBase = d * slabBytes + slabOff;
#pragma unroll
    for (int i = 0; i < F4_PER_T; ++i) {
      const unsigned f4   = (unsigned)(tid + i * THREADS);
      const unsigned gOff = dBase + f4 * 16u;
      const unsigned lOff = ldsBase + f4 * 16u;
      asm volatile("global_store_async_from_lds_b128 %0, %1, %2 th:TH_STORE_NT"
                   :
                   : "v"(gOff), "v"(lOff), "s"(out)
                   : "memory");
    }
  }
  // Drain outstanding async stores before wave exit.
  asm volatile("s_wait_asynccnt 0" ::: "memory");
}

// ---------------------------------------------------------------------------
// Kernel B: out[d, 0, :] = inputs[d, :]  (8 MiB coalesced copy, NT stores)
// ---------------------------------------------------------------------------
__global__ __launch_bounds__(256) void slot0_copy_kernel(
    const float* __restrict__ inputs, float* __restrict__ out) {
  const int idx = blockIdx.x * 256 + threadIdx.x;   // float4 index, row = idx/256
  const int d = idx >> 8;                           // 256 float4 per D=1024 row
  const int c = idx & 255;

  const v4f* src = (const v4f*)inputs;
  v4f* dst = (v4f*)out;
  const v4f v = src[idx];
  __builtin_nontemporal_store(v, &dst[(size_t)d * (L_WIN * D_FEAT / 4) + c]);
}

// ---------------------------------------------------------------------------
extern "C" void kernel_launch(void* const* d_in, const int* in_sizes, int n_in,
                              void* d_out, int out_size, void* d_ws,
                              size_t ws_size, hipStream_t stream) {
  (void)in_sizes; (void)n_in; (void)out_size; (void)d_ws; (void)ws_size;

  const float* inputs = (const float*)d_in[0];  // [T, D] f32
  const float* memory = (const float*)d_in[1];  // [L, D] f32
  float* out = (float*)d_out;                   // [T, L, D] f32

  // Kernel A: 7 tiles x (2048/16) timestep-groups = 896 blocks.
  dim3 gridA(N_TILES * (T_STEPS / TB));
  tail_broadcast_kernel<<<gridA, THREADS, 0, stream>>>(memory, out);

  // Kernel B: 2048*1024/4 float4 / 256 threads = 2048 blocks.
  dim3 gridB((T_STEPS * D_FEAT / 4) / 256);
  slot0_copy_kernel<<<gridB, 256, 0, stream>>>(inputs, out);
}